// Quantizer_85160611545366
// MI455X (gfx1250) — compile-verified
//
#include <hip/hip_runtime.h>
#include <hip/hip_bf16.h>

#define IC       4096
#define THREADS  256
#define ELEMS    16      // IC / THREADS
#define NWAVES   8       // THREADS / 32 (wave32 on gfx1250)

__device__ __forceinline__ float wave_sum(float v) {
#pragma unroll
  for (int off = 16; off > 0; off >>= 1) v += __shfl_xor(v, off, 32);
  return v;
}
__device__ __forceinline__ float wave_min(float v) {
#pragma unroll
  for (int off = 16; off > 0; off >>= 1) v = fminf(v, __shfl_xor(v, off, 32));
  return v;
}
__device__ __forceinline__ float wave_max(float v) {
#pragma unroll
  for (int off = 16; off > 0; off >>= 1) v = fmaxf(v, __shfl_xor(v, off, 32));
  return v;
}

// One block per output row. Row is staged global->LDS with the CDNA5 async
// copy path (GLOBAL_LOAD_ASYNC_TO_LDS_B128, tracked by ASYNCcnt), then held
// in VGPRs for all 101 candidate evaluations.
__global__ __launch_bounds__(THREADS)
void fitquant_kernel(const float* __restrict__ w,
                     const int* __restrict__ bitw,
                     float* __restrict__ out) {
  __shared__ __align__(16) float srow[IC];
  __shared__ float sred[NWAVES], smin[NWAVES], smax[NWAVES];

  const int row  = blockIdx.x;
  const int tid  = threadIdx.x;
  const int lane = tid & 31;
  const int wid  = tid >> 5;

  // ---- async stage: each thread copies its own 64B chunk into LDS ----
  const float* gsrc = w + (size_t)row * IC + (size_t)tid * ELEMS;
  unsigned long long gaddr = (unsigned long long)gsrc;
  unsigned lds_off = (unsigned)(unsigned long long)&srow[tid * ELEMS];

  asm volatile(
      "global_load_async_to_lds_b128 %0, %1, off\n\t"
      "global_load_async_to_lds_b128 %0, %1, off offset:16\n\t"
      "global_load_async_to_lds_b128 %0, %1, off offset:32\n\t"
      "global_load_async_to_lds_b128 %0, %1, off offset:48"
      :: "v"(lds_off), "v"(gaddr) : "memory");
  asm volatile("s_wait_asynccnt 0" ::: "memory");

  float wv[ELEMS];
  {
    const float4* s4 = (const float4*)&srow[tid * ELEMS];
#pragma unroll
    for (int i = 0; i < 4; ++i) {
      float4 t = s4[i];
      wv[4 * i + 0] = t.x; wv[4 * i + 1] = t.y;
      wv[4 * i + 2] = t.z; wv[4 * i + 3] = t.w;
    }
  }

  const int   bw    = bitw[0];
  const float maxq  = (float)((1 << bw) - 1);
  const float zero0 = (maxq + 1.0f) * 0.5f;
  const float qlo   = -zero0;          // clip(q,0,maxq)-zero0 lower bound
  const float qhi   = maxq - zero0;    // upper bound

  // ---- row min/max (seeded with 0 => implements min(.,0)/max(.,0)) ----
  float lmin = 0.0f, lmax = 0.0f;
#pragma unroll
  for (int i = 0; i < ELEMS; ++i) {
    lmin = fminf(lmin, wv[i]);
    lmax = fmaxf(lmax, wv[i]);
  }
  lmin = wave_min(lmin);
  lmax = wave_max(lmax);
  if (lane == 0) { smin[wid] = lmin; smax[wid] = lmax; }
  __syncthreads();
  float xmin = 0.0f, xmax = 0.0f;
#pragma unroll
  for (int i = 0; i < NWAVES; ++i) {
    xmin = fminf(xmin, smin[i]);
    xmax = fmaxf(xmax, smax[i]);
  }
  if (xmin == 0.0f && xmax == 0.0f) { xmin = -1.0f; xmax = 1.0f; }
  const float rng = xmax - xmin;

  // ---- 101-candidate grid search (same ordering as the reference scan) ----
  float best      = __builtin_inff();
  float bestScale = rng / maxq;

  for (int c = 0; c < 101; ++c) {
    float p = (c == 0)  ? 1.0f
            : (c <= 50) ? 1.0f + 0.002f * (float)c
                        : 1.0f - 0.002f * (float)(c - 50);
    const float scale1 = p * rng / maxq;
    const float inv    = 1.0f / scale1;

    float local = 0.0f;
#pragma unroll
    for (int i = 0; i < ELEMS; ++i) {
      float r  = rintf(wv[i] * inv);           // round half-to-even (jnp.round)
      float qm = fminf(fmaxf(r, qlo), qhi);    // clip to [0,maxq], zero removed
      float d  = fmaf(scale1, qm, -wv[i]);     // dq - w
      // |d|^2.4 via raw hw transcendentals; log2(0)=-inf -> exp2(-inf)=0.
      float e  = __builtin_amdgcn_exp2f(2.4f * __builtin_amdgcn_logf(fabsf(d)));
      local += e;
    }

    // NOTE: with LAMBDA==1.0, err_ns + err_s == full-row sum(e); the
    // saliency mask (and thus diag(Hinv1)) cancels, so Hinv1 is never read.
    local = wave_sum(local);
    if (lane == 0) sred[wid] = local;
    __syncthreads();
    float err = 0.0f;
#pragma unroll
    for (int i = 0; i < NWAVES; ++i) err += sred[i];
    __syncthreads();

    if (err < best) { best = err; bestScale = scale1; }
  }

  // ---- final quantize with fitted scale ----
  const float inv = 1.0f / bestScale;
  float4* o4 = (float4*)(out + (size_t)row * IC + (size_t)tid * ELEMS);
#pragma unroll
  for (int i = 0; i < 4; ++i) {
    float4 t;
    float* tp = &t.x;
#pragma unroll
    for (int j = 0; j < 4; ++j) {
      float x  = wv[4 * i + j];
      float r  = rintf(x * inv);
      float qm = fminf(fmaxf(r, qlo), qhi);
      tp[j] = bestScale * qm;
    }
    o4[i] = t;
  }
}

extern "C" void kernel_launch(void* const* d_in, const int* in_sizes, int n_in,
                              void* d_out, int out_size, void* d_ws, size_t ws_size,
                              hipStream_t stream) {
  const float* w  = (const float*)d_in[0];
  // d_in[1] (Hinv1) is mathematically unused: only its diagonal feeds the
  // saliency mask, and with lambda == 1.0 the masked sums recombine into the
  // full-row sum. Skipping it saves a 64 MB read.
  const int*   bw = (const int*)d_in[2];
  float*       out = (float*)d_out;

  const int oc = out_size / IC;   // 4096 rows
  fitquant_kernel<<<dim3(oc), dim3(THREADS), 0, stream>>>(w, bw, out);
}